// MPNNs_10763188043968
// MI455X (gfx1250) — compile-verified
//
#include <hip/hip_runtime.h>
#include <stdint.h>

#define HF      128
#define OUTC    64
#define BN_EPS  1e-5f
#define LDW     132   // padded LDS row stride (128 would alias all 64 banks row-to-row)
#define LDWP    68    // padded stride for 64-wide Wp tile
#define ROWS    64    // rows of X per workgroup in the dual GEMM

typedef __attribute__((ext_vector_type(2))) float v2f;
typedef __attribute__((ext_vector_type(8))) float v8f;

#if defined(__gfx1250__) && __has_builtin(__builtin_amdgcn_tensor_load_to_lds)
#define HAVE_TDM 1
#else
#define HAVE_TDM 0
#endif

#if HAVE_TDM
typedef __attribute__((ext_vector_type(4))) unsigned int v4u;
typedef __attribute__((ext_vector_type(8))) int          v8i;
typedef __attribute__((ext_vector_type(4))) int          v4i;

// TDM copy of a 128x128 fp32 row-major matrix into LDS with stride-132 padding.
// pad: after every 128 DWORDs (pad_interval=6) insert 4 DWORDs (pad_amount=3).
__device__ __forceinline__ void tdm_load_w128(const float* gsrc, unsigned int ldsAddr) {
    unsigned long long ga = (unsigned long long)(uintptr_t)gsrc;
    v4u g0;
    g0.x = 1u;                                            // count=1, user descriptor
    g0.y = ldsAddr;                                       // LDS byte address
    g0.z = (unsigned int)(ga & 0xFFFFFFFFu);              // global_addr[31:0]
    g0.w = (unsigned int)((ga >> 32) & 0x1FFFFFFu) | (2u << 30); // global_addr[56:32] | type=2
    v8i g1;
    g1[0] = (int)((2u << 16) | (1u << 20) | (6u << 22) | (3u << 25)); // 4B elems, pad en, 128dw/4dw
    g1[1] = (int)(128u << 16);  // tensor_dim0[15:0] = 128
    g1[2] = (int)(128u << 16);  // tensor_dim0 hi=0 | tensor_dim1[15:0]=128
    g1[3] = (int)(128u << 16);  // tensor_dim1 hi=0 | tile_dim0=128
    g1[4] = 128;                // tile_dim1=128, tile_dim2=0 (2D)
    g1[5] = 128;                // tensor_dim0_stride = 128
    g1[6] = 0;
    g1[7] = 0;
    v4i z4 = {0, 0, 0, 0};
    v8i z8 = {0, 0, 0, 0, 0, 0, 0, 0};
    __builtin_amdgcn_tensor_load_to_lds(g0, g1, z4, z4, z8, 0);   // clang-23 6-arg form
}
#endif

// ---------------------------------------------------------------- degree prep
__global__ __launch_bounds__(256) void k_init_deg(float* __restrict__ deg, int N) {
    int i = blockIdx.x * 256 + threadIdx.x;
    if (i < N) deg[i] = 1.0f;                       // self-loop
}

__global__ __launch_bounds__(256) void k_count_deg(const long long* __restrict__ dst,
                                                   float* __restrict__ deg, int E) {
    int e = blockIdx.x * 256 + threadIdx.x;
    if (e < E) unsafeAtomicAdd(&deg[(int)dst[e]], 1.0f);
}

__global__ __launch_bounds__(256) void k_dinv(float* __restrict__ deg, int N) {
    int i = blockIdx.x * 256 + threadIdx.x;
    if (i < N) deg[i] = __frsqrt_rn(deg[i]);        // deg >= 1 always (self-loops)
}

// ------------------------------------------------- dual GEMM: h=x@Wg, cv=x@Wl+bl+bg
// Block = 64 rows of x; 8 waves own 16-col tiles and sweep 4 row tiles over
// LDS-resident weights (staged once per block via TDM when available).
__global__ __launch_bounds__(256) void k_dual_gemm(
    const float* __restrict__ X,  const float* __restrict__ Wg,
    const float* __restrict__ Wl, const float* __restrict__ bg,
    const float* __restrict__ bl, float* __restrict__ Hb,
    float* __restrict__ Cv, int N)
{
    __shared__ float sWg[HF * LDW];
    __shared__ float sWl[HF * LDW];
    __shared__ float sX[ROWS * LDW];

    const int tid     = threadIdx.x;
    const int rowBase = blockIdx.x * ROWS;

#if HAVE_TDM
    if (tid < 32) {                                 // wave 0 only: TDM ignores EXEC
        tdm_load_w128(Wg, (unsigned int)(uintptr_t)&sWg[0]);
        tdm_load_w128(Wl, (unsigned int)(uintptr_t)&sWl[0]);
        __builtin_amdgcn_s_wait_tensorcnt(0);
    }
#else
    for (int i = tid; i < HF * HF / 4; i += 256) {
        int r = i >> 5, c4 = (i & 31) * 4;
        float4 g = ((const float4*)(Wg + r * HF))[i & 31];
        float4 l = ((const float4*)(Wl + r * HF))[i & 31];
        float* dg = &sWg[r * LDW + c4];
        dg[0] = g.x; dg[1] = g.y; dg[2] = g.z; dg[3] = g.w;
        float* dl = &sWl[r * LDW + c4];
        dl[0] = l.x; dl[1] = l.y; dl[2] = l.z; dl[3] = l.w;
    }
#endif
    // stage ROWSx128 X tile (guarded at the ragged tail)
    for (int i = tid; i < ROWS * HF / 4; i += 256) {
        int r = i >> 5, c4 = (i & 31) * 4;
        int row = rowBase + r;
        float4 v = make_float4(0.f, 0.f, 0.f, 0.f);
        if (row < N) v = ((const float4*)(X + (size_t)row * HF))[i & 31];
        float* dx = &sX[r * LDW + c4];
        dx[0] = v.x; dx[1] = v.y; dx[2] = v.z; dx[3] = v.w;
    }
    __syncthreads();

    const int lane = tid & 31;
    const int wid  = tid >> 5;
    const int col  = wid * 16;
    const int mrow = lane & 15;          // A: M index / B: N index
    const int koff = (lane >> 4) * 2;    // lanes 16-31 carry K+2,K+3
    const int mAdd = (lane >> 4) * 8;    // C/D: lanes 16-31 hold rows M+8
    const int n    = lane & 15;
    const float bsum = bl[col + n] + bg[col + n];

    for (int rt = 0; rt < ROWS; rt += 16) {
        v8f cg = {}; v8f cl = {};
        for (int k0 = 0; k0 < HF; k0 += 4) {
            v2f a, bgv, blv;
            a.x   = sX[(rt + mrow) * LDW + k0 + koff];
            a.y   = sX[(rt + mrow) * LDW + k0 + koff + 1];
            bgv.x = sWg[(k0 + koff)     * LDW + col + mrow];
            bgv.y = sWg[(k0 + koff + 1) * LDW + col + mrow];
            blv.x = sWl[(k0 + koff)     * LDW + col + mrow];
            blv.y = sWl[(k0 + koff + 1) * LDW + col + mrow];
            cg = __builtin_amdgcn_wmma_f32_16x16x4_f32(false, a, false, bgv, (short)0, cg, false, false);
            cl = __builtin_amdgcn_wmma_f32_16x16x4_f32(false, a, false, blv, (short)0, cl, false, false);
        }
#pragma unroll
        for (int j = 0; j < 8; ++j) {
            int row = rowBase + rt + j + mAdd;
            if (row < N) {
                Hb[(size_t)row * HF + col + n] = cg[j];
                Cv[(size_t)row * HF + col + n] = cl[j] + bsum;
            }
        }
    }
}

// ---------------------------------------------------------- edge scatter-add
// One wave per edge; lane handles 4 features (float4 gather + 4 f32 atomics, L2-resident).
__global__ __launch_bounds__(256) void k_scatter(
    const long long* __restrict__ src, const long long* __restrict__ dst,
    const float* __restrict__ dinv, const float* __restrict__ Hb,
    float* __restrict__ Cv, int E)
{
    int e    = blockIdx.x * 8 + (threadIdx.x >> 5);
    int lane = threadIdx.x & 31;
    if (e >= E) return;
    int s = (int)src[e], d = (int)dst[e];
    float w = dinv[s] * dinv[d];
    float4 v = ((const float4*)(Hb + (size_t)s * HF))[lane];
    float* o = Cv + (size_t)d * HF + lane * 4;
    unsafeAtomicAdd(o + 0, w * v.x);
    unsafeAtomicAdd(o + 1, w * v.y);
    unsafeAtomicAdd(o + 2, w * v.z);
    unsafeAtomicAdd(o + 3, w * v.w);
}

// ------------------------------------------- self-loop + BN(eval) + ReLU fuse
__global__ __launch_bounds__(256) void k_bnrelu(
    const float* __restrict__ Cv, const float* __restrict__ Hb,
    const float* __restrict__ dinv,
    const float* __restrict__ g,  const float* __restrict__ b,
    const float* __restrict__ rm, const float* __restrict__ rv,
    float* __restrict__ Xo, int N)
{
    int t = blockIdx.x * 256 + threadIdx.x;       // one thread per 4 features
    if (t >= N * 32) return;
    int v  = t >> 5;
    int q  = t & 31;
    int f4 = q * 4;
    float di = dinv[v];
    float w  = di * di;                           // self-loop norm
    float4 hv = ((const float4*)(Hb + (size_t)v * HF))[q];
    float4 cv = ((const float4*)(Cv + (size_t)v * HF))[q];
    float hs[4] = {hv.x, hv.y, hv.z, hv.w};
    float cs[4] = {cv.x, cv.y, cv.z, cv.w};
    float os[4];
#pragma unroll
    for (int j = 0; j < 4; ++j) {
        int f   = f4 + j;
        float y = cs[j] + w * hs[j];
        y = (y - rm[f]) * __frsqrt_rn(rv[f] + BN_EPS) * g[f] + b[f];
        os[j] = fmaxf(y, 0.0f);
    }
    ((float4*)(Xo + (size_t)v * HF))[q] = make_float4(os[0], os[1], os[2], os[3]);
}

// ------------------------------------------------------ final: out = x@Wp + bp
// Block = 32 rows; 8 waves cover 2 row-tiles x 4 col-tiles of 16x16.
__global__ __launch_bounds__(256) void k_out_gemm(
    const float* __restrict__ X, const float* __restrict__ Wp,
    const float* __restrict__ bp, float* __restrict__ out, int N)
{
    __shared__ float sW[HF * LDWP];
    __shared__ float sX[32 * LDW];

    const int tid     = threadIdx.x;
    const int rowBase = blockIdx.x * 32;

    for (int i = tid; i < HF * OUTC / 4; i += 256) {
        int r = i >> 4, c4 = (i & 15) * 4;
        float4 v = ((const float4*)(Wp + r * OUTC))[i & 15];
        float* dw = &sW[r * LDWP + c4];
        dw[0] = v.x; dw[1] = v.y; dw[2] = v.z; dw[3] = v.w;
    }
    for (int i = tid; i < 32 * HF / 4; i += 256) {
        int r = i >> 5, c4 = (i & 31) * 4;
        int row = rowBase + r;
        float4 v = make_float4(0.f, 0.f, 0.f, 0.f);
        if (row < N) v = ((const float4*)(X + (size_t)row * HF))[i & 31];
        float* dx = &sX[r * LDW + c4];
        dx[0] = v.x; dx[1] = v.y; dx[2] = v.z; dx[3] = v.w;
    }
    __syncthreads();

    const int lane = tid & 31;
    const int wid  = tid >> 5;
    const int rt   = (wid >> 2) * 16;
    const int col  = (wid & 3) * 16;
    const int mrow = lane & 15;
    const int koff = (lane >> 4) * 2;

    v8f c = {};
    for (int k0 = 0; k0 < HF; k0 += 4) {
        v2f a, bv;
        a.x  = sX[(rt + mrow) * LDW + k0 + koff];
        a.y  = sX[(rt + mrow) * LDW + k0 + koff + 1];
        bv.x = sW[(k0 + koff)     * LDWP + col + mrow];
        bv.y = sW[(k0 + koff + 1) * LDWP + col + mrow];
        c = __builtin_amdgcn_wmma_f32_16x16x4_f32(false, a, false, bv, (short)0, c, false, false);
    }

    const int mAdd = (lane >> 4) * 8;
    const int n    = lane & 15;
    const float bb = bp[col + n];
#pragma unroll
    for (int j = 0; j < 8; ++j) {
        int row = rowBase + rt + j + mAdd;
        if (row < N) out[(size_t)row * OUTC + col + n] = c[j] + bb;
    }
}

// ----------------------------------------------------------------- launcher
extern "C" void kernel_launch(void* const* d_in, const int* in_sizes, int n_in,
                              void* d_out, int out_size, void* d_ws, size_t ws_size,
                              hipStream_t stream) {
    const float*     x     = (const float*)d_in[0];
    const long long* ei    = (const long long*)d_in[1];
    const float*     Wg    = (const float*)d_in[2];
    const float*     bg    = (const float*)d_in[3];
    const float*     Wl    = (const float*)d_in[4];
    const float*     bl    = (const float*)d_in[5];
    const float*     bn_g  = (const float*)d_in[6];
    const float*     bn_b  = (const float*)d_in[7];
    const float*     bn_rm = (const float*)d_in[8];
    const float*     bn_rv = (const float*)d_in[9];
    const float*     Wp    = (const float*)d_in[10];
    const float*     bp    = (const float*)d_in[11];
    float*           out   = (float*)d_out;

    const int N = in_sizes[0] / HF;
    const int E = in_sizes[1] / 2;
    const long long* esrc = ei;
    const long long* edst = ei + E;

    // workspace carve-up: dinv | Hb | Cv | X0 | X1
    float* ws    = (float*)d_ws;
    size_t NH    = (size_t)N * HF;
    float* dinv  = ws;
    size_t off   = ((size_t)N + 255) & ~(size_t)255;
    float* Hb    = ws + off;
    float* Cv    = Hb + NH;
    float* X0    = Cv + NH;
    float* X1    = X0 + NH;

    // degree / normalization coefficients
    k_init_deg<<<(N + 255) / 256, 256, 0, stream>>>(dinv, N);
    k_count_deg<<<(E + 255) / 256, 256, 0, stream>>>(edst, dinv, E);
    k_dinv<<<(N + 255) / 256, 256, 0, stream>>>(dinv, N);

    const float* xin  = x;
    float*       xout = X0;
    const int gBlocks = (N + ROWS - 1) / ROWS;
    for (int l = 0; l < 3; ++l) {
        k_dual_gemm<<<gBlocks, 256, 0, stream>>>(xin, Wg + l * HF * HF, Wl + l * HF * HF,
                                                 bg + l * HF, bl + l * HF, Hb, Cv, N);
        k_scatter<<<(E + 7) / 8, 256, 0, stream>>>(esrc, edst, dinv, Hb, Cv, E);
        k_bnrelu<<<(N * 32 + 255) / 256, 256, 0, stream>>>(Cv, Hb, dinv,
                                                           bn_g + l * HF, bn_b + l * HF,
                                                           bn_rm + l * HF, bn_rv + l * HF,
                                                           xout, N);
        xin  = xout;
        xout = (l == 0) ? X1 : X0;
    }
    k_out_gemm<<<(N + 31) / 32, 256, 0, stream>>>(xin, Wp, bp, out, N);
}